// HierarchicalClusterLayer_46926812676802
// MI455X (gfx1250) — compile-verified
//
#include <hip/hip_runtime.h>
#include <math.h>

#define DIM    768
#define HEADS  4
#define HD     3072      // HEADS*DIM
#define BATCH  128
#define NUM_L1 16
#define NUM_L2 64
#define OUTW   (NUM_L1 + NUM_L2)

typedef __attribute__((ext_vector_type(2))) float v2f;
typedef __attribute__((ext_vector_type(8))) float v8f;

// ---------------------------------------------------------------------------
// C[m,n] = sum_{k<768} A[zA + (m0+m)*lda + k] * W[zB + (n0+n)*ldb + k] (+ bias)
// One wave computes a 16x64 block of C (4 N-tiles) via V_WMMA_F32_16X16X4_F32.
// A fragment is reused across 4 WMMAs; the next k-step's 5 fragments are
// software-prefetched before the current step's WMMAs so loads overlap
// matrix issue (avoids s_wait_loadcnt 0x0 in front of every WMMA).
// blockIdx.x -> M tile (16), blockIdx.y -> N block (64), blockIdx.z -> head.
// All N dims used here (128/768/3072) are multiples of 64.
// ---------------------------------------------------------------------------
__global__ __launch_bounds__(32)
void gemm_nt_f32_wmma(const float* __restrict__ A,
                      const float* __restrict__ W,
                      float* __restrict__ C,
                      const float* __restrict__ bias,
                      int lda, int ldb, int ldc,
                      long strideAz, long strideBz, long strideCz)
{
    const int lane = threadIdx.x;          // 0..31, wave32
    const int r    = lane & 15;            // row within fragment
    const int kh   = (lane >> 4) << 1;     // lanes 0-15 -> k {0,1}; 16-31 -> k {2,3}
    const long m0  = (long)blockIdx.x * 16;
    const long n0  = (long)blockIdx.y * 64;

    const float* aRow = A + (long)blockIdx.z * strideAz + (m0 + r) * (long)lda + kh;
    const float* wRowBase = W + (long)blockIdx.z * strideBz + (n0 + r) * (long)ldb + kh;
    const float* w0 = wRowBase;
    const float* w1 = wRowBase + 16 * (long)ldb;
    const float* w2 = wRowBase + 32 * (long)ldb;
    const float* w3 = wRowBase + 48 * (long)ldb;

    v8f acc0 = {}, acc1 = {}, acc2 = {}, acc3 = {};

    // prologue: fetch k-step 0
    v2f av = *(const v2f*)(aRow);
    v2f b0 = *(const v2f*)(w0);
    v2f b1 = *(const v2f*)(w1);
    v2f b2 = *(const v2f*)(w2);
    v2f b3 = *(const v2f*)(w3);

#pragma unroll 4
    for (int k = 4; k < DIM; k += 4) {
        // prefetch next k-step while current WMMAs issue
        v2f avn = *(const v2f*)(aRow + k);
        v2f b0n = *(const v2f*)(w0 + k);
        v2f b1n = *(const v2f*)(w1 + k);
        v2f b2n = *(const v2f*)(w2 + k);
        v2f b3n = *(const v2f*)(w3 + k);
        acc0 = __builtin_amdgcn_wmma_f32_16x16x4_f32(false, av, false, b0, (short)0, acc0, false, false);
        acc1 = __builtin_amdgcn_wmma_f32_16x16x4_f32(false, av, false, b1, (short)0, acc1, false, false);
        acc2 = __builtin_amdgcn_wmma_f32_16x16x4_f32(false, av, false, b2, (short)0, acc2, false, false);
        acc3 = __builtin_amdgcn_wmma_f32_16x16x4_f32(false, av, false, b3, (short)0, acc3, false, false);
        av = avn; b0 = b0n; b1 = b1n; b2 = b2n; b3 = b3n;
    }
    // epilogue k-step
    acc0 = __builtin_amdgcn_wmma_f32_16x16x4_f32(false, av, false, b0, (short)0, acc0, false, false);
    acc1 = __builtin_amdgcn_wmma_f32_16x16x4_f32(false, av, false, b1, (short)0, acc1, false, false);
    acc2 = __builtin_amdgcn_wmma_f32_16x16x4_f32(false, av, false, b2, (short)0, acc2, false, false);
    acc3 = __builtin_amdgcn_wmma_f32_16x16x4_f32(false, av, false, b3, (short)0, acc3, false, false);

    // C/D layout: VGPR j holds M=j (lanes 0-15) and M=j+8 (lanes 16-31); N=lane&15.
    const int  col   = r;
    const int  rbase = (lane >> 4) * 8;
    float* cRow = C + (long)blockIdx.z * strideCz + n0 + col;
    const v8f accs[4] = { acc0, acc1, acc2, acc3 };
#pragma unroll
    for (int g = 0; g < 4; ++g) {
        const float bval = bias ? bias[n0 + g * 16 + col] : 0.0f;
#pragma unroll
        for (int j = 0; j < 8; ++j)
            cRow[(m0 + rbase + j) * (long)ldc + g * 16] = accs[g][j] + bval;
    }
}

// ---------------------------------------------------------------------------
// s00[t*H + h] = dot(Q[t, h*768 .. ], K[t, h*768 .. ]) ; one wave per (t,h)
// ---------------------------------------------------------------------------
__global__ __launch_bounds__(32)
void diag_dot_kernel(const float* __restrict__ Q, const float* __restrict__ K,
                     float* __restrict__ s00)
{
    const int th = blockIdx.x;
    const int t  = th / HEADS;
    const int h  = th % HEADS;
    const float* q = Q + (long)t * HD + h * DIM;
    const float* k = K + (long)t * HD + h * DIM;
    float sum = 0.0f;
    for (int i = threadIdx.x; i < DIM; i += 32) sum += q[i] * k[i];
#pragma unroll
    for (int o = 16; o; o >>= 1) sum += __shfl_down(sum, o);
    if (threadIdx.x == 0) s00[th] = sum;
}

// ---------------------------------------------------------------------------
// block-wide sum (256 threads = 8 waves)
// ---------------------------------------------------------------------------
__device__ inline float block_sum(float v, float* red)
{
    const int lane = threadIdx.x & 31;
    const int wid  = threadIdx.x >> 5;
#pragma unroll
    for (int o = 16; o; o >>= 1) v += __shfl_down(v, o);
    if (lane == 0) red[wid] = v;
    __syncthreads();
    if (wid == 0) {
        float r = (lane < 8) ? red[lane] : 0.0f;
#pragma unroll
        for (int o = 4; o; o >>= 1) r += __shfl_down(r, o);
        if (lane == 0) red[8] = r;
    }
    __syncthreads();
    float r = red[8];
    __syncthreads();
    return r;
}

// ---------------------------------------------------------------------------
// Per (b,t): softmax over {s00, s01}, mix P vectors + residual, LN, LN, score.
// grid = (T, BATCH), block = 256 (each thread owns 3 of the 768 channels)
// ---------------------------------------------------------------------------
__global__ __launch_bounds__(256)
void epilogue_score(const float* __restrict__ s00,   // [T*HEADS]
                    const float* __restrict__ s01,   // [HEADS][T][BATCH]
                    const float* __restrict__ Ptok,  // [T][HEADS*768]
                    const float* __restrict__ Ptp,   // [BATCH][HEADS*768]
                    const float* __restrict__ fcb,
                    const float* __restrict__ tokens,// [T][768]
                    const float* __restrict__ lng, const float* __restrict__ lnb,
                    const float* __restrict__ slng, const float* __restrict__ slnb,
                    const float* __restrict__ sw,  const float* __restrict__ sbias,
                    float* __restrict__ out, int T, int outOff)
{
    __shared__ float red[9];
    const int t = blockIdx.x;
    const int b = blockIdx.y;
    const float scale = 1.0f / sqrtf((float)DIM);

    float a0[HEADS], a1[HEADS];
#pragma unroll
    for (int h = 0; h < HEADS; ++h) {
        const float sc0 = s00[t * HEADS + h] * scale;
        const float sc1 = s01[((long)h * T + t) * BATCH + b] * scale;
        const float m  = fmaxf(sc0, sc1);
        const float e0 = expf(sc0 - m);
        const float e1 = expf(sc1 - m);
        const float inv = 1.0f / (e0 + e1);
        a0[h] = e0 * inv;
        a1[h] = e1 * inv;
    }

    float y[3];
    float s = 0.0f, sq = 0.0f;
#pragma unroll
    for (int i = 0; i < 3; ++i) {
        const int e = threadIdx.x + i * 256;
        float v = fcb[e] + tokens[(long)t * DIM + e];
#pragma unroll
        for (int h = 0; h < HEADS; ++h) {
            v += a0[h] * Ptok[(long)t * HD + h * DIM + e]
               + a1[h] * Ptp [(long)b * HD + h * DIM + e];
        }
        y[i] = v; s += v; sq += v * v;
    }
    s  = block_sum(s,  red);
    sq = block_sum(sq, red);
    float mean = s * (1.0f / DIM);
    float var  = sq * (1.0f / DIM) - mean * mean;
    float rstd = rsqrtf(var + 1e-5f);

    float z[3];
    float s2 = 0.0f, sq2 = 0.0f;
#pragma unroll
    for (int i = 0; i < 3; ++i) {
        const int e = threadIdx.x + i * 256;
        const float zz = (y[i] - mean) * rstd * lng[e] + lnb[e];
        z[i] = zz; s2 += zz; sq2 += zz * zz;
    }
    s2  = block_sum(s2,  red);
    sq2 = block_sum(sq2, red);
    float mean2 = s2 * (1.0f / DIM);
    float var2  = sq2 * (1.0f / DIM) - mean2 * mean2;
    float rstd2 = rsqrtf(var2 + 1e-5f);

    float dot = 0.0f;
#pragma unroll
    for (int i = 0; i < 3; ++i) {
        const int e = threadIdx.x + i * 256;
        const float w = (z[i] - mean2) * rstd2 * slng[e] + slnb[e];
        dot += w * sw[e];
    }
    dot = block_sum(dot, red);
    if (threadIdx.x == 0)
        out[(long)b * OUTW + outOff + t] = dot + sbias[0];
}

// ---------------------------------------------------------------------------
extern "C" void kernel_launch(void* const* d_in, const int* in_sizes, int n_in,
                              void* d_out, int out_size, void* d_ws, size_t ws_size,
                              hipStream_t stream)
{
    (void)in_sizes; (void)n_in; (void)out_size; (void)ws_size;

    const float* task_emb    = (const float*)d_in[0];
    const float* tokens_lv[2]= { (const float*)d_in[1], (const float*)d_in[2] };
    const float* task_proj_w = (const float*)d_in[3];
    const float* task_proj_b = (const float*)d_in[4];

    // per-level parameter tables (level 0 = l1 @ idx 5.., level 1 = l2 @ idx 16..)
    const int base[2] = { 5, 16 };
    const int Tlv [2] = { NUM_L1, NUM_L2 };
    const int off [2] = { 0, NUM_L1 };

    float* ws     = (float*)d_ws;
    float* task_p = ws;                      // 128*768
    float* Qt     = task_p + BATCH * DIM;    // 64*3072 (max T)
    float* Kt     = Qt   + NUM_L2 * HD;
    float* Vt     = Kt   + NUM_L2 * HD;
    float* Qp     = Vt   + NUM_L2 * HD;      // 128*3072
    float* Kp     = Qp   + BATCH * HD;
    float* Vp     = Kp   + BATCH * HD;
    float* Ptok   = Vp   + BATCH * HD;       // 64*3072
    float* Ptp    = Ptok + NUM_L2 * HD;      // 128*3072
    float* s00    = Ptp  + BATCH * HD;       // 64*4
    float* s01    = s00  + NUM_L2 * HEADS;   // 4*64*128

    dim3 wave(32);

    // task_p = task_emb @ task_proj_w.T + task_proj_b   (128 x 768, K=768)
    gemm_nt_f32_wmma<<<dim3(BATCH / 16, DIM / 64, 1), wave, 0, stream>>>(
        task_emb, task_proj_w, task_p, task_proj_b, DIM, DIM, DIM, 0, 0, 0);

    for (int L = 0; L < 2; ++L) {
        const int   T   = Tlv[L];
        const int   ib  = base[L];
        const float* wq   = (const float*)d_in[ib + 0];
        const float* wk   = (const float*)d_in[ib + 1];
        const float* wv   = (const float*)d_in[ib + 2];
        const float* fcw  = (const float*)d_in[ib + 3];
        const float* fcb  = (const float*)d_in[ib + 4];
        const float* lng  = (const float*)d_in[ib + 5];
        const float* lnb  = (const float*)d_in[ib + 6];
        const float* slng = (const float*)d_in[ib + 7];
        const float* slnb = (const float*)d_in[ib + 8];
        const float* sw   = (const float*)d_in[ib + 9];
        const float* sb   = (const float*)d_in[ib + 10];
        const float* tok  = tokens_lv[L];

        // token projections: (T x 3072) = tokens (T x 768) @ w.T
        gemm_nt_f32_wmma<<<dim3(T / 16, HD / 64, 1), wave, 0, stream>>>(
            tok, wq, Qt, nullptr, DIM, DIM, HD, 0, 0, 0);
        gemm_nt_f32_wmma<<<dim3(T / 16, HD / 64, 1), wave, 0, stream>>>(
            tok, wk, Kt, nullptr, DIM, DIM, HD, 0, 0, 0);
        gemm_nt_f32_wmma<<<dim3(T / 16, HD / 64, 1), wave, 0, stream>>>(
            tok, wv, Vt, nullptr, DIM, DIM, HD, 0, 0, 0);

        // task projections: (128 x 3072)
        gemm_nt_f32_wmma<<<dim3(BATCH / 16, HD / 64, 1), wave, 0, stream>>>(
            task_p, wq, Qp, nullptr, DIM, DIM, HD, 0, 0, 0);
        gemm_nt_f32_wmma<<<dim3(BATCH / 16, HD / 64, 1), wave, 0, stream>>>(
            task_p, wk, Kp, nullptr, DIM, DIM, HD, 0, 0, 0);
        gemm_nt_f32_wmma<<<dim3(BATCH / 16, HD / 64, 1), wave, 0, stream>>>(
            task_p, wv, Vp, nullptr, DIM, DIM, HD, 0, 0, 0);

        // per-head fc fold: P[x,h,:] = V[x,h,:] @ fcw_h.T   (z = head)
        gemm_nt_f32_wmma<<<dim3(T / 16, DIM / 64, HEADS), wave, 0, stream>>>(
            Vt, fcw, Ptok, nullptr, HD, HD, HD, DIM, DIM, DIM);
        gemm_nt_f32_wmma<<<dim3(BATCH / 16, DIM / 64, HEADS), wave, 0, stream>>>(
            Vp, fcw, Ptp, nullptr, HD, HD, HD, DIM, DIM, DIM);

        // cross scores: s01[h][t][b] = Qt[t,h,:] . Kp[b,h,:]
        gemm_nt_f32_wmma<<<dim3(T / 16, BATCH / 64, HEADS), wave, 0, stream>>>(
            Qt, Kp, s01, nullptr, HD, HD, BATCH, DIM, DIM, (long)T * BATCH);

        // self scores: s00[t,h] = Qt[t,h,:] . Kt[t,h,:]
        diag_dot_kernel<<<dim3(T * HEADS), wave, 0, stream>>>(Qt, Kt, s00);

        // fused softmax + mix + residual + LN + LN + score
        epilogue_score<<<dim3(T, BATCH), dim3(256), 0, stream>>>(
            s00, s01, Ptok, Ptp, fcb, tok, lng, lnb, slng, slnb, sw, sb,
            (float*)d_out, T, off[L]);
    }
}